// SPDecoder_33689723470055
// MI455X (gfx1250) — compile-verified
//
#include <hip/hip_runtime.h>

// 1-D multi-scale deformable attention sampling for MI455X (gfx1250, wave32).
// bs=4, H=8, D=32, nq=4096, LEVELS=(16384,8192,4096,2048), P=8.

#define BS 4
#define NQ 4096
#define NH 8
#define ND 32
#define NK 30720                 // sum(LEVELS)
#define JOBS (BS * NQ * NH)      // 131072 (one wave32 each)
#define WAVES_PER_BLOCK 8
#define THREADS (WAVES_PER_BLOCK * 32)

typedef __attribute__((ext_vector_type(2))) int v2i_t;
typedef v2i_t __attribute__((address_space(1)))* gbl_v2i_ptr;
typedef v2i_t __attribute__((address_space(3)))* lds_v2i_ptr;

__global__ __launch_bounds__(THREADS) void msda1d_kernel(
    const float* __restrict__ value,   // (bs, NK, H, D)
    const float* __restrict__ loc,     // (bs, nq, H, Lv, P, 1)
    const float* __restrict__ wgt,     // (bs, nq, H, Lv, P)
    float* __restrict__ out)           // (bs, nq, H*D)
{
    const unsigned tid  = threadIdx.x;
    const unsigned lane = tid & 31u;
    const unsigned wave = tid >> 5;
    const unsigned job  = blockIdx.x * WAVES_PER_BLOCK + wave; // (b*NQ+q)*H + h
    const unsigned h    = job & 7u;          // H == 8
    const unsigned b    = (job >> 3) >> 12;  // NQ == 4096

    // Per-wave staging area: [0..31] = locations, [32..63] = weights.
    __shared__ float lds[WAVES_PER_BLOCK * 64];
    float* myLds = &lds[wave * 64];

    const size_t pbase = (size_t)job * 32;   // loc/wgt are contiguous per job

    float myloc, myw;
#if defined(__gfx1250__) && __has_builtin(__builtin_amdgcn_global_load_async_to_lds_b64)
    {
        // lanes 0..15 stage 2 floats of loc each, lanes 16..31 stage 2 floats of wgt;
        // LDS float offset 2*lane lays them out as [loc(32) | wgt(32)].
        const float* g = (lane < 16u) ? (loc + pbase + 2u * lane)
                                      : (wgt + pbase + 2u * (lane - 16u));
        float* l = myLds + 2u * lane;
        __builtin_amdgcn_global_load_async_to_lds_b64(
            (gbl_v2i_ptr)g, (lds_v2i_ptr)l, /*offset=*/0, /*cpol=*/0);
        asm volatile("s_wait_asynccnt 0" ::: "memory");
        myloc = myLds[lane];
        myw   = myLds[32 + lane];
    }
#else
    (void)myLds;
    myloc = loc[pbase + lane];
    myw   = wgt[pbase + lane];
#endif

    // Lane 'lane' owns sample point lp = lane: lvl = lp>>3, p = lp&7.
    const int L   = 16384 >> (lane >> 3);
    const int off = 32768 - 2 * L;           // level key offset: 0,16384,24576,28672

    float x  = fmaf(myloc, (float)L, -1.0f);
    float xf = floorf(x);
    float lx = x - xf;
    float hx = 1.0f - lx;
    int   xl = (int)xf;
    int   xh = xl + 1;

    float cl = (xl >= 0 && xl < L) ? (myw * hx) : 0.0f;  // weight * hx * m_low
    float ch = (xh >= 0 && xh < L) ? (myw * lx) : 0.0f;  // weight * lx * m_high
    int il = xl < 0 ? 0 : (xl > L - 1 ? L - 1 : xl);
    int ih = xh < 0 ? 0 : (xh > L - 1 ? L - 1 : xh);

    // Element offset of value row (b fixed): (off+idx)*H*D + h*D
    const int hoff = (int)(h << 5);
    int ol = ((off + il) << 8) + hoff;
    int oh = ((off + ih) << 8) + hoff;
    int cli = __builtin_bit_cast(int, cl);
    int chi = __builtin_bit_cast(int, ch);

    const float* __restrict__ valB = value + (size_t)b * (NK * NH * ND);

    float acc = 0.0f;
#pragma unroll
    for (int pt = 0; pt < 32; ++pt) {
        // Broadcast point pt's row offsets + fused coefficients to SGPRs.
        int   sol = __builtin_amdgcn_readlane(ol, pt);
        int   soh = __builtin_amdgcn_readlane(oh, pt);
        float scl = __builtin_bit_cast(float, __builtin_amdgcn_readlane(cli, pt));
        float sch = __builtin_bit_cast(float, __builtin_amdgcn_readlane(chi, pt));
        // Two coalesced 128B row loads (one float per lane), two FMAs.
        acc = fmaf(scl, valB[(size_t)sol + lane], acc);
        acc = fmaf(sch, valB[(size_t)soh + lane], acc);
    }

    // out[(b*NQ+q)*H*D + h*D + d] == out[job*32 + lane]
    out[(size_t)job * ND + lane] = acc;
}

extern "C" void kernel_launch(void* const* d_in, const int* in_sizes, int n_in,
                              void* d_out, int out_size, void* d_ws, size_t ws_size,
                              hipStream_t stream) {
    (void)in_sizes; (void)n_in; (void)d_ws; (void)ws_size; (void)out_size;
    const float* value = (const float*)d_in[0];
    const float* loc   = (const float*)d_in[1];
    const float* wgt   = (const float*)d_in[2];
    float*       out   = (float*)d_out;

    dim3 grid(JOBS / WAVES_PER_BLOCK);   // 16384 blocks
    dim3 block(THREADS);                 // 256 threads = 8 wave32
    msda1d_kernel<<<grid, block, 0, stream>>>(value, loc, wgt, out);
}